// FLNO1d_30004641530304
// MI455X (gfx1250) — compile-verified
//
#include <hip/hip_runtime.h>
#include <math.h>

// Problem constants
#define NFFT   4096
#define NSTAGE 12
#define Wd     32
#define Bd     32
#define Md     8
#define BW     1024          // Bd*Wd sequences / frequency-tile rows*cols
#define FC1d   128

typedef __attribute__((ext_vector_type(2))) float v2f;
typedef __attribute__((ext_vector_type(8))) float v8f;

__device__ __forceinline__ float gelu_exact(float v) {
    return 0.5f * v * (1.0f + erff(v * 0.7071067811865475f));
}

// D = A(16x4 f32) * B(4x16 f32) + C(16x16 f32), fp32 WMMA (CDNA5)
__device__ __forceinline__ v8f wmma4(v2f a, v2f b, v8f c) {
    return __builtin_amdgcn_wmma_f32_16x16x4_f32(
        false, a, false, b, (short)0, c, false, false);
}

// ---------------------------------------------------------------------------
// fc0 + gelu:  h[b][w][n] = gelu(x[b][n]*fc0_w[0][w] + t[n]*fc0_w[1][w] + b[w])
// ---------------------------------------------------------------------------
__global__ void __launch_bounds__(256)
fc0_kernel(const float* __restrict__ x, const float* __restrict__ fw,
           const float* __restrict__ fb, float* __restrict__ h)
{
    int gid = blockIdx.x * 256 + threadIdx.x;        // b*W*N + w*N + n
    int n  = gid & (NFFT - 1);
    int bw = gid >> 12;
    int w  = bw & (Wd - 1);
    int b  = bw >> 5;
    float t = (float)n * (1.0f / 4095.0f);
    float v = x[b * NFFT + n] * fw[w] + t * fw[Wd + w] + fb[w];
    h[gid] = gelu_exact(v);
}

// ---------------------------------------------------------------------------
// Per-layer frequency-domain operator:
//   G[x][i][o] = Hw[o][i][x] + conv_w[o][i] + spec(x<8) + conj-mirror(x>N-8)
//   Hw[o][i][x] = sum_m residue[o][i][m] / (i*w_x - pole[o][i][m])
// ---------------------------------------------------------------------------
__global__ void __launch_bounds__(256)
build_G_kernel(const float2* __restrict__ pole, const float2* __restrict__ res,
               const float2* __restrict__ spec, const float* __restrict__ convw,
               float2* __restrict__ G)
{
    int gid = blockIdx.x * 256 + threadIdx.x;        // x*1024 + i*32 + o
    int io = gid & (BW - 1);
    int x  = gid >> 10;
    int i  = io >> 5;
    int o  = io & (Wd - 1);

    float kk = (x < NFFT / 2) ? (float)x : (float)(x - NFFT);
    float w  = 6.283185307179586f * kk * (4095.0f / 4096.0f);   // 2*pi*fftfreq

    const float2* pp = pole + (o * Wd + i) * Md;
    const float2* rr = res  + (o * Wd + i) * Md;
    float sr = 0.f, si = 0.f;
#pragma unroll
    for (int m = 0; m < Md; ++m) {
        float dr = -pp[m].x;
        float di = w - pp[m].y;
        float inv = 1.0f / (dr * dr + di * di);
        sr += (rr[m].x * dr + rr[m].y * di) * inv;
        si += (rr[m].y * dr - rr[m].x * di) * inv;
    }
    sr += convw[o * Wd + i];
    if (x < Md) {                                   // spectral-conv modes 0..7
        float2 s = spec[(i * Wd + o) * Md + x];
        sr += s.x; si += s.y;
    } else if (x > NFFT - Md) {                     // hermitian mirror (conj)
        float2 s = spec[(i * Wd + o) * Md + (NFFT - x)];
        sr += s.x; si -= s.y;
    }
    G[(size_t)x * BW + i * Wd + o] = make_float2(sr, si);
}

// ---------------------------------------------------------------------------
// Radix-2 Stockham autosort FFT on LDS (length 4096, 256 threads).
// After 12 stages (even), result lands back in X.
// ---------------------------------------------------------------------------
__device__ __forceinline__ void fft_lds(float2* X, float2* Y, float sign, int tid)
{
    int l = 1;
    float2* src = X;
    float2* dst = Y;
#pragma unroll 1
    for (int s = 0; s < NSTAGE; ++s) {
        for (int j = tid; j < NFFT / 2; j += 256) {
            int k = j & (l - 1);
            float ang = sign * 3.14159265358979323f * (float)k / (float)l;
            float sn, cs;
            __sincosf(ang, &sn, &cs);
            float2 a = src[j];
            float2 b = src[j + NFFT / 2];
            float2 bw = make_float2(b.x * cs - b.y * sn, b.x * sn + b.y * cs);
            int j2 = 2 * j - k;                      // (j/l)*2l + k
            dst[j2]     = make_float2(a.x + bw.x, a.y + bw.y);
            dst[j2 + l] = make_float2(a.x - bw.x, a.y - bw.y);
        }
        __syncthreads();
        float2* t = src; src = dst; dst = t;
        l <<= 1;
    }
}

// Forward FFT of real h[seq][n]; writes frequency-major alphaT[x][seq].
__global__ void __launch_bounds__(256)
fft_fwd_kernel(const float* __restrict__ h, float2* __restrict__ alphaT)
{
    __shared__ float2 smem[2 * NFFT];                // 64 KB of 320 KB WGP LDS
    const int seq = blockIdx.x;                      // b*32 + w
    const float* src = h + (size_t)seq * NFFT;
    for (int i = threadIdx.x; i < NFFT; i += 256)
        smem[i] = make_float2(src[i], 0.0f);
    __syncthreads();
    fft_lds(smem, smem + NFFT, -1.0f, threadIdx.x);
    for (int i = threadIdx.x; i < NFFT; i += 256)
        alphaT[(size_t)i * BW + seq] = smem[i];
}

// ---------------------------------------------------------------------------
// Per-frequency batched complex GEMM via V_WMMA_F32_16X16X4_F32.
// One wave per frequency: S[b][o] = sum_i A[b][i] * G[i][o]  (32x32x32 complex)
// In-place: AT tile is fully consumed before the stores.
// ---------------------------------------------------------------------------
__global__ void __launch_bounds__(256)
freq_gemm_kernel(float2* __restrict__ AT, const float2* __restrict__ G)
{
    const int wave = threadIdx.x >> 5;
    const int lane = threadIdx.x & 31;
    const int x = (blockIdx.x << 3) + wave;
    float2* A        = AT + (size_t)x * BW;
    const float2* Gx = G  + (size_t)x * BW;

    const int mrow  = lane & 15;   // M (A rows / D cols) within a 16-tile
    const int khalf = lane >> 4;   // selects K pair {0,1} vs {2,3}

    v8f cr00 = {}, cr01 = {}, cr10 = {}, cr11 = {};
    v8f ci00 = {}, ci01 = {}, ci10 = {}, ci11 = {};

#pragma unroll
    for (int kc = 0; kc < 8; ++kc) {
        const int kb = kc * 4 + khalf * 2;
        // A fragments (16x4): lane holds (M=mrow[+16], K=kb..kb+1)
        float2 a0c0 = A[mrow * Wd + kb];
        float2 a0c1 = A[mrow * Wd + kb + 1];
        float2 a1c0 = A[(mrow + 16) * Wd + kb];
        float2 a1c1 = A[(mrow + 16) * Wd + kb + 1];
        v2f a0r = {a0c0.x, a0c1.x}, a0i = {a0c0.y, a0c1.y};
        v2f a1r = {a1c0.x, a1c1.x}, a1i = {a1c0.y, a1c1.y};
        // B fragments (4x16): lane holds (K=kb..kb+1, N=mrow[+16])
        float2 g0c0 = Gx[kb * Wd + mrow];
        float2 g0c1 = Gx[(kb + 1) * Wd + mrow];
        float2 g1c0 = Gx[kb * Wd + mrow + 16];
        float2 g1c1 = Gx[(kb + 1) * Wd + mrow + 16];
        v2f g0r = {g0c0.x, g0c1.x}, g0i = {g0c0.y, g0c1.y};
        v2f g1r = {g1c0.x, g1c1.x}, g1i = {g1c0.y, g1c1.y};
        v2f g0in = -g0i, g1in = -g1i;   // f32 WMMA has no A/B NEG modifier

        // Sr = Ar*Gr - Ai*Gi ; Si = Ar*Gi + Ai*Gr
        cr00 = wmma4(a0r, g0r, cr00); cr00 = wmma4(a0i, g0in, cr00);
        ci00 = wmma4(a0r, g0i, ci00); ci00 = wmma4(a0i, g0r,  ci00);
        cr01 = wmma4(a0r, g1r, cr01); cr01 = wmma4(a0i, g1in, cr01);
        ci01 = wmma4(a0r, g1i, ci01); ci01 = wmma4(a0i, g1r,  ci01);
        cr10 = wmma4(a1r, g0r, cr10); cr10 = wmma4(a1i, g0in, cr10);
        ci10 = wmma4(a1r, g0i, ci10); ci10 = wmma4(a1i, g0r,  ci10);
        cr11 = wmma4(a1r, g1r, cr11); cr11 = wmma4(a1i, g1in, cr11);
        ci11 = wmma4(a1r, g1i, ci11); ci11 = wmma4(a1i, g1r,  ci11);
    }

    // D layout (16x16 f32): VGPR r -> (M = r + 8*khalf, N = mrow)
#pragma unroll
    for (int r = 0; r < 8; ++r) {
        int row = r + khalf * 8;
        A[row * Wd + mrow]             = make_float2(cr00[r], ci00[r]);
        A[row * Wd + mrow + 16]        = make_float2(cr01[r], ci01[r]);
        A[(row + 16) * Wd + mrow]      = make_float2(cr10[r], ci10[r]);
        A[(row + 16) * Wd + mrow + 16] = make_float2(cr11[r], ci11[r]);
    }
}

// ---------------------------------------------------------------------------
// Inverse FFT (+1, 1/N) fused with conv bias + gelu -> new h[seq][n]
// ---------------------------------------------------------------------------
__global__ void __launch_bounds__(256)
ifft_gelu_kernel(const float2* __restrict__ ST, const float* __restrict__ convb,
                 float* __restrict__ hout)
{
    __shared__ float2 smem[2 * NFFT];
    const int seq = blockIdx.x;                      // b*32 + o
    const int o = seq & (Wd - 1);
    for (int i = threadIdx.x; i < NFFT; i += 256)
        smem[i] = ST[(size_t)i * BW + seq];
    __syncthreads();
    fft_lds(smem, smem + NFFT, 1.0f, threadIdx.x);
    const float bias = convb[o];
    const float invn = 1.0f / (float)NFFT;
    float* dst = hout + (size_t)seq * NFFT;
    for (int i = threadIdx.x; i < NFFT; i += 256)
        dst[i] = gelu_exact(smem[i].x * invn + bias);
}

// ---------------------------------------------------------------------------
// Head via WMMA: y[b][n] = fc2_b + sum_f fc2_w[f] *
//                          gelu(fc1_b[f] + sum_w h[b][w][n]*fc1_w[w][f])
// One wave per 16-row block of (b,n); fc1 = [16,32]x[32,128] in 64 WMMAs;
// epilogue fuses bias+gelu+fc2 with a 16-lane shfl_xor reduction (valid for
// the 16x16 f32 D layout: M = r + 8*(lane/16), N = lane%16).
// ---------------------------------------------------------------------------
__global__ void __launch_bounds__(256)
head_kernel(const float* __restrict__ h, const float* __restrict__ w1,
            const float* __restrict__ b1, const float* __restrict__ w2,
            const float* __restrict__ b2, float* __restrict__ out)
{
    __shared__ float lw1[Wd * FC1d];                 // 16 KB: fc1_w per block
    for (int i = threadIdx.x; i < Wd * FC1d; i += 256)
        lw1[i] = w1[i];
    __syncthreads();

    const int wave  = threadIdx.x >> 5;
    const int lane  = threadIdx.x & 31;
    const int mrow  = lane & 15;
    const int khalf = lane >> 4;

    const int rb = blockIdx.x * 8 + wave;            // 16-row block of (b,n)
    const int b  = rb >> 8;                          // N/16 = 256 blocks per b
    const int n0 = (rb & 255) << 4;

    v8f acc[8] = {};                                 // 8 N-tiles of FC1
#pragma unroll
    for (int kc = 0; kc < 8; ++kc) {
        const int k0 = kc * 4 + khalf * 2;           // fc1 input channel (w)
        v2f a;                                       // A: (M=n0+mrow, K=k0..k0+1)
        a[0] = h[((size_t)b * Wd + k0) * NFFT + n0 + mrow];
        a[1] = h[((size_t)b * Wd + k0 + 1) * NFFT + n0 + mrow];
#pragma unroll
        for (int tn = 0; tn < 8; ++tn) {
            v2f bf;                                  // B: (K=k0..k0+1, N)
            bf[0] = lw1[k0 * FC1d + tn * 16 + mrow];
            bf[1] = lw1[(k0 + 1) * FC1d + tn * 16 + mrow];
            acc[tn] = wmma4(a, bf, acc[tn]);
        }
    }

    // Epilogue: f = tn*16 + mrow ; row n = n0 + r + 8*khalf
    float part[8];
#pragma unroll
    for (int r = 0; r < 8; ++r) part[r] = 0.0f;
#pragma unroll
    for (int tn = 0; tn < 8; ++tn) {
        const float b1v = b1[tn * 16 + mrow];
        const float w2v = w2[tn * 16 + mrow];
#pragma unroll
        for (int r = 0; r < 8; ++r)
            part[r] += gelu_exact(acc[tn][r] + b1v) * w2v;
    }
#pragma unroll
    for (int r = 0; r < 8; ++r) {
#pragma unroll
        for (int off = 1; off < 16; off <<= 1)
            part[r] += __shfl_xor(part[r], off, 32);
    }
    if (mrow == 0) {
        const float bias2 = b2[0];
        float* dst = out + (size_t)b * NFFT + n0 + khalf * 8;
#pragma unroll
        for (int r = 0; r < 8; ++r)
            dst[r] = part[r] + bias2;
    }
}

// ---------------------------------------------------------------------------
extern "C" void kernel_launch(void* const* d_in, const int* in_sizes, int n_in,
                              void* d_out, int out_size, void* d_ws, size_t ws_size,
                              hipStream_t stream)
{
    const float*  x       = (const float*)d_in[0];
    const float*  fc0_w   = (const float*)d_in[1];
    const float*  fc0_b   = (const float*)d_in[2];
    const float2* pole    = (const float2*)d_in[3];
    const float2* residue = (const float2*)d_in[4];
    const float2* spec_w  = (const float2*)d_in[5];
    const float*  conv_w  = (const float*)d_in[6];
    const float*  conv_b  = (const float*)d_in[7];
    const float*  fc1_w   = (const float*)d_in[8];
    const float*  fc1_b   = (const float*)d_in[9];
    const float*  fc2_w   = (const float*)d_in[10];
    const float*  fc2_b   = (const float*)d_in[11];
    float* out = (float*)d_out;

    // Workspace layout (96 MB total):
    //   hA, hB   : [B*W*N] f32  (16 MB each)
    //   alphaT   : [N][B*W] c64 (32 MB)  -- also reused in-place as S
    //   G        : [N][W*W] c64 (32 MB)
    char* ws = (char*)d_ws;
    const size_t HBYTES = (size_t)Bd * Wd * NFFT * sizeof(float);
    float*  hA     = (float*)(ws);
    float*  hB     = (float*)(ws + HBYTES);
    float2* alphaT = (float2*)(ws + 2 * HBYTES);
    float2* G      = (float2*)(ws + 2 * HBYTES + (size_t)NFFT * BW * sizeof(float2));

    fc0_kernel<<<(Bd * Wd * NFFT) / 256, 256, 0, stream>>>(x, fc0_w, fc0_b, hA);

    for (int l = 0; l < 4; ++l) {
        const float* hin  = (l & 1) ? hB : hA;
        float*       hout = (l & 1) ? hA : hB;
        build_G_kernel<<<(NFFT * BW) / 256, 256, 0, stream>>>(
            pole + (size_t)l * Wd * Wd * Md, residue + (size_t)l * Wd * Wd * Md,
            spec_w + (size_t)l * Wd * Wd * Md, conv_w + (size_t)l * Wd * Wd, G);
        fft_fwd_kernel<<<BW, 256, 0, stream>>>(hin, alphaT);
        freq_gemm_kernel<<<NFFT / 8, 256, 0, stream>>>(alphaT, G);
        ifft_gelu_kernel<<<BW, 256, 0, stream>>>(alphaT, conv_b + l * Wd, hout);
    }

    head_kernel<<<(Bd * NFFT) / 16 / 8, 256, 0, stream>>>(hA, fc1_w, fc1_b,
                                                          fc2_w, fc2_b, out);
}